// WaveRNN_26096221291277
// MI455X (gfx1250) — compile-verified
//
#include <hip/hip_runtime.h>
#include <math.h>

typedef __attribute__((ext_vector_type(2))) float v2f;
typedef __attribute__((ext_vector_type(8))) float v8f;

#define RNN_H 5

__device__ __forceinline__ v8f wmma4(v2f a, v2f b, v8f c) {
  // D(16x16 f32) = A(16x4 f32) * B(4x16 f32) + C
  return __builtin_amdgcn_wmma_f32_16x16x4_f32(false, a, false, b, (short)0, c,
                                               false, false);
}

__device__ __forceinline__ float tanh_fast(float x) {
#if __has_builtin(__builtin_amdgcn_tanhf)
  return __builtin_amdgcn_tanhf(x);            // v_tanh_f32 on gfx1250
#elif __has_builtin(__builtin_amdgcn_tanh_f32)
  return __builtin_amdgcn_tanh_f32(x);
#else
  return tanhf(x);
#endif
}

__global__ __launch_bounds__(32) void wavernn_fused(
    const float* __restrict__ hidden, const float* __restrict__ X,
    const float* __restrict__ Wih,    const float* __restrict__ Whh,
    const float* __restrict__ bih,    const float* __restrict__ bhh,
    const float* __restrict__ fc1w,   const float* __restrict__ fc1b,
    const float* __restrict__ fc2w,   const float* __restrict__ fc2b,
    float* __restrict__ out, int B, int T)
{
  const int  lane = threadIdx.x & 31;
  const int  j    = lane & 15;          // M index (hidden row) for A / batch col for B
  const bool lo   = lane < 16;          // lanes 0-15 carry K=0,1 of A; 16-31 carry K=2,3 (zeroed)
  const int  base = blockIdx.x * 16;
  const bool bval = (base + j) < B;
  const int  b    = bval ? (base + j) : (B - 1);

  // ---- A operands: W_hh and fc1_w, rows j>=5 and the K=2,3 half zeroed ----
  const bool jv = lo && (j < RNN_H);
  v2f Ar01 = { jv ? Whh[j * RNN_H + 0] : 0.f, jv ? Whh[j * RNN_H + 1] : 0.f };
  v2f Ar23 = { jv ? Whh[j * RNN_H + 2] : 0.f, jv ? Whh[j * RNN_H + 3] : 0.f };
  v2f Ar4  = { jv ? Whh[j * RNN_H + 4] : 0.f, 0.f };
  v2f Af01 = { jv ? fc1w[j * RNN_H + 0] : 0.f, jv ? fc1w[j * RNN_H + 1] : 0.f };
  v2f Af23 = { jv ? fc1w[j * RNN_H + 2] : 0.f, jv ? fc1w[j * RNN_H + 3] : 0.f };
  v2f Af4  = { jv ? fc1w[j * RNN_H + 4] : 0.f, 0.f };

  // ---- uniform scalars (compiler scalarizes these to s_loads) ----
  const float wi0 = Wih[0], wi1 = Wih[1], wi2 = Wih[2], wi3 = Wih[3], wi4 = Wih[4];
  const float bi0 = bih[0] + bhh[0], bi1 = bih[1] + bhh[1], bi2 = bih[2] + bhh[2],
              bi3 = bih[3] + bhh[3], bi4 = bih[4] + bhh[4];
  const float f20 = fc2w[0], f21 = fc2w[1], f22 = fc2w[2], f23 = fc2w[3],
              f24 = fc2w[4], f2c = fc2b[0];
  const v8f Cf = { fc1b[0], fc1b[1], fc1b[2], fc1b[3], fc1b[4], 0.f, 0.f, 0.f };

  // ---- hidden state, transposed: register h_k holds h[batch=lane, k] ----
  float h0 = hidden[b * RNN_H + 0];
  float h1 = hidden[b * RNN_H + 1];
  float h2 = hidden[b * RNN_H + 2];
  float h3 = hidden[b * RNN_H + 3];
  float h4 = hidden[b * RNN_H + 4];

  const float* __restrict__ Xr = X   + (size_t)b * (size_t)T;
  float*       __restrict__ Yr = out + (size_t)b * (size_t)T;

  // One RNN step + head: 6x v_wmma_f32_16x16x4_f32, 5x tanh, ~15 VALU.
  auto step = [&](float x) -> float {
    v8f c = { fmaf(x, wi0, bi0), fmaf(x, wi1, bi1), fmaf(x, wi2, bi2),
              fmaf(x, wi3, bi3), fmaf(x, wi4, bi4), 0.f, 0.f, 0.f };
    v2f B01 = { h0, h1 }, B23 = { h2, h3 }, B4 = { h4, 0.f };
    v8f d = wmma4(Ar01, B01, c);
    d = wmma4(Ar23, B23, d);
    d = wmma4(Ar4,  B4,  d);
    h0 = tanh_fast(d[0]); h1 = tanh_fast(d[1]); h2 = tanh_fast(d[2]);
    h3 = tanh_fast(d[3]); h4 = tanh_fast(d[4]);
    v2f N01 = { h0, h1 }, N23 = { h2, h3 }, N4 = { h4, 0.f };
    v8f e = wmma4(Af01, N01, Cf);
    e = wmma4(Af23, N23, e);
    e = wmma4(Af4,  N4,  e);
    float y = f2c;
    y = fmaf(f20, fmaxf(e[0], 0.f), y);
    y = fmaf(f21, fmaxf(e[1], 0.f), y);
    y = fmaf(f22, fmaxf(e[2], 0.f), y);
    y = fmaf(f23, fmaxf(e[3], 0.f), y);
    y = fmaf(f24, fmaxf(e[4], 0.f), y);
    return y;
  };

  auto do4 = [&](const float4& xv, int tb) {
    float4 y4;
    y4.x = step(xv.x); y4.y = step(xv.y); y4.z = step(xv.z); y4.w = step(xv.w);
    if (lo && bval) *(float4*)(Yr + tb) = y4;   // predicated store only; WMMA stays full-EXEC
  };

  const int Tm = ((T & 3) == 0) ? (T & ~31) : 0;  // vector path needs 16B alignment

  if (Tm > 0) {
    float4 xa0 = *(const float4*)(Xr + 0);
    float4 xa1 = *(const float4*)(Xr + 4);
    float4 xa2 = *(const float4*)(Xr + 8);
    float4 xa3 = *(const float4*)(Xr + 12);
    for (int t0 = 0; t0 < Tm; t0 += 32) {
      if (t0 + 96 < T) __builtin_prefetch(Xr + t0 + 96, 0, 1);
      // prefetch chunk B (t0+16 .. t0+31) into registers
      const int tb = t0 + 16;  // tb + 16 <= Tm always
      float4 xb0 = *(const float4*)(Xr + tb + 0);
      float4 xb1 = *(const float4*)(Xr + tb + 4);
      float4 xb2 = *(const float4*)(Xr + tb + 8);
      float4 xb3 = *(const float4*)(Xr + tb + 12);
      // process chunk A while chunk B is in flight
      do4(xa0, t0 + 0); do4(xa1, t0 + 4); do4(xa2, t0 + 8); do4(xa3, t0 + 12);
      // prefetch next chunk A (t0+32), clamped so the last (unused) load stays in-bounds
      const int ta = (t0 + 32 + 16 <= T) ? (t0 + 32) : 0;
      xa0 = *(const float4*)(Xr + ta + 0);
      xa1 = *(const float4*)(Xr + ta + 4);
      xa2 = *(const float4*)(Xr + ta + 8);
      xa3 = *(const float4*)(Xr + ta + 12);
      // process chunk B
      do4(xb0, tb + 0); do4(xb1, tb + 4); do4(xb2, tb + 8); do4(xb3, tb + 12);
    }
  }
  // scalar tail (dead for T=2048)
  for (int t = Tm; t < T; ++t) {
    float y = step(Xr[t]);
    if (lo && bval) Yr[t] = y;
  }
}

extern "C" void kernel_launch(void* const* d_in, const int* in_sizes, int n_in,
                              void* d_out, int out_size, void* d_ws, size_t ws_size,
                              hipStream_t stream) {
  (void)n_in; (void)out_size; (void)d_ws; (void)ws_size;
  const float* hidden = (const float*)d_in[0];
  const float* X      = (const float*)d_in[1];
  const float* Wih    = (const float*)d_in[2];
  const float* Whh    = (const float*)d_in[3];
  const float* bih    = (const float*)d_in[4];
  const float* bhh    = (const float*)d_in[5];
  const float* f1w    = (const float*)d_in[6];
  const float* f1b    = (const float*)d_in[7];
  const float* f2w    = (const float*)d_in[8];
  const float* f2b    = (const float*)d_in[9];
  float* out = (float*)d_out;

  const int Bn = in_sizes[0] / RNN_H;        // hidden is [1, B, H]
  const int T  = in_sizes[1] / Bn;           // X is [B, T, 1]
  const int grid = (Bn + 15) / 16;           // 16 batch rows per wave32

  wavernn_fused<<<grid, 32, 0, stream>>>(hidden, X, Wih, Whh, bih, bhh,
                                         f1w, f1b, f2w, f2b, out, Bn, T);
}